// GCN_5334349382408
// MI455X (gfx1250) — compile-verified
//
#include <hip/hip_runtime.h>
#include <cstdint>

// ---------------------------------------------------------------------------
// GCN (3-layer) for MI455X / gfx1250.
//  - fp32 WMMA (V_WMMA_F32_16X16X4_F32) for the dense h = x @ W GEMMs.
//  - CSR built on device once per call; gather-based aggregation (no float
//    atomics in the hot loop), fused norm/self-loop/bias/relu.
// edge_index is passed by the harness as int32 ("integer -> const int*").
// ---------------------------------------------------------------------------

typedef __attribute__((ext_vector_type(2))) float v2f;
typedef __attribute__((ext_vector_type(8))) float v8f;

#define KDIM    128          // inner dimension (all layers)
#define BM      128          // rows per GEMM block (8 waves x 16 rows)
#define KCHUNK  64           // K-chunk staged in LDS
#define WSTRIDE (KCHUNK + 2) // padded LDS stride (even -> 8B-aligned b64 reads)
#define SCAN_B  1024

// ------------------------------ GEMM (WMMA) --------------------------------
// H[M, NOUT] = X[M,128] @ W[128, NOUT].  One 256-thread block covers 128 rows;
// wave w computes rows [blk*128 + 16w, +16).  W is staged transposed in LDS.
template <int NOUT>
__global__ __launch_bounds__(256) void gemm_wmma_f32(
    const float* __restrict__ X, const float* __restrict__ W,
    float* __restrict__ H, int M) {
  constexpr int NT  = (NOUT + 15) / 16;   // 8 (N=128) or 3 (N=40, padded 48)
  constexpr int NPAD = NT * 16;
  __shared__ float Wt[NPAD][WSTRIDE];     // transposed W chunk: Wt[n][k]

  const int tid  = threadIdx.x;
  const int wave = tid >> 5;
  const int lane = tid & 31;
  const int m    = lane & 15;             // A row / B col within tile
  const int koff = (lane >> 4) << 1;      // 0 (lanes 0-15) or 2 (lanes 16-31)
  const int rowBase = blockIdx.x * BM + wave * 16;

  int row = rowBase + m;
  if (row >= M) row = M - 1;              // clamp (stores are predicated)
  const float* xrow = X + (long long)row * KDIM;

  v8f zero = {0.f, 0.f, 0.f, 0.f, 0.f, 0.f, 0.f, 0.f};
  v8f acc[NT];
#pragma unroll
  for (int t = 0; t < NT; ++t) acc[t] = zero;

  for (int kc = 0; kc < KDIM; kc += KCHUNK) {
    // Cooperative transposed load of W[kc..kc+63][*] into LDS (zero-pad cols).
    for (int idx = tid; idx < KCHUNK * NPAD; idx += 256) {
      int kk = idx / NPAD;
      int nn = idx - kk * NPAD;
      Wt[nn][kk] = (nn < NOUT) ? W[(long long)(kc + kk) * NOUT + nn] : 0.f;
    }
    __syncthreads();

    for (int k0 = 0; k0 < KCHUNK; k0 += 4) {
      // A fragment: X[m][kc+k0+koff .. +1]  (global_load_b64, 8B aligned)
      v2f a = *(const v2f*)(xrow + kc + k0 + koff);
#pragma unroll
      for (int t = 0; t < NT; ++t) {
        const int n = t * 16 + m;
        // B fragment: Wt[n][k0+koff .. +1]  (ds_load_b64, 8B aligned)
        v2f b = *(const v2f*)(&Wt[n][k0 + koff]);
        acc[t] = __builtin_amdgcn_wmma_f32_16x16x4_f32(
            false, a, false, b, (short)0, acc[t], false, false);
      }
    }
    __syncthreads();
  }

  // C/D layout: VGPR v -> row M = v + 8*(lane>=16), col N = lane&15.
  const int half = lane >> 4;
#pragma unroll
  for (int t = 0; t < NT; ++t) {
    const int n = t * 16 + m;
    if (n < NOUT) {
#pragma unroll
      for (int v = 0; v < 8; ++v) {
        const int r = rowBase + v + 8 * half;
        if (r < M) H[(long long)r * NOUT + n] = acc[t][v];
      }
    }
  }
}

// --------------------------- CSR construction ------------------------------
__global__ __launch_bounds__(256) void zero2_i32(int* a, int* b, int N_) {
  int i = blockIdx.x * blockDim.x + threadIdx.x;
  if (i < N_) { a[i] = 0; b[i] = 0; }
}

__global__ __launch_bounds__(256) void hist_deg(const int* __restrict__ dst,
                                                int* __restrict__ degI, int E_) {
  int e = blockIdx.x * blockDim.x + threadIdx.x;
  if (e < E_) atomicAdd(&degI[dst[e]], 1);
}

__global__ __launch_bounds__(256) void calc_dinv(const int* __restrict__ degI,
                                                 float* __restrict__ dinv, int N_) {
  int i = blockIdx.x * blockDim.x + threadIdx.x;
  if (i < N_) dinv[i] = rsqrtf((float)degI[i] + 1.0f);
}

__global__ __launch_bounds__(SCAN_B) void scan_block(
    const int* __restrict__ degI, int* __restrict__ offs,
    int* __restrict__ parts, int N_) {
  __shared__ int sh[SCAN_B];
  const int tid = threadIdx.x;
  const int i = blockIdx.x * SCAN_B + tid;
  const int v = (i < N_) ? degI[i] : 0;
  sh[tid] = v;
  __syncthreads();
  for (int off = 1; off < SCAN_B; off <<= 1) {   // Hillis-Steele inclusive
    int t = (tid >= off) ? sh[tid - off] : 0;
    __syncthreads();
    sh[tid] += t;
    __syncthreads();
  }
  if (i < N_) offs[i] = sh[tid] - v;             // exclusive within block
  if (tid == SCAN_B - 1) parts[blockIdx.x] = sh[tid];
}

__global__ void scan_partials(int* parts, int nb) {
  if (blockIdx.x == 0 && threadIdx.x == 0) {
    int run = 0;
    for (int b = 0; b < nb; ++b) { int v = parts[b]; parts[b] = run; run += v; }
  }
}

__global__ __launch_bounds__(256) void scan_add(int* __restrict__ offs,
                                                const int* __restrict__ parts,
                                                int N_, int E_) {
  int i = blockIdx.x * blockDim.x + threadIdx.x;
  if (i < N_) offs[i] += parts[i / SCAN_B];
  if (i == 0) offs[N_] = E_;
}

__global__ __launch_bounds__(256) void csr_fill(
    const int* __restrict__ src, const int* __restrict__ dst,
    const int* __restrict__ offs, int* __restrict__ cursor,
    const float* __restrict__ dinv, int* __restrict__ csr_s,
    float* __restrict__ csr_w, int E_) {
  int e = blockIdx.x * blockDim.x + threadIdx.x;
  if (e >= E_) return;
  const int d = dst[e];
  const int s = src[e];
  const int pos = atomicAdd(&cursor[d], 1);
  const int idx = offs[d] + pos;
  csr_s[idx] = s;
  csr_w[idx] = dinv[s];
}

// --------------------------- Aggregation (gather) --------------------------
// One wave per node, float4 per lane (128 feats). out_i = d_i*(sum + d_i*h_i)+b.
__global__ __launch_bounds__(256) void gcn_agg128(
    const float* __restrict__ H, const float* __restrict__ bias,
    const float* __restrict__ dinv, const int* __restrict__ offs,
    const int* __restrict__ csr_s, const float* __restrict__ csr_w,
    float* __restrict__ out, int N_, int do_relu) {
  const int wid  = (int)((blockIdx.x * blockDim.x + threadIdx.x) >> 5);
  const int lane = threadIdx.x & 31;
  if (wid >= N_) return;
  const float di = dinv[wid];
  float4 hv = ((const float4*)(H + (long long)wid * 128))[lane];
  float4 acc = make_float4(di * hv.x, di * hv.y, di * hv.z, di * hv.w);
  const int e1 = offs[wid + 1];
  for (int e = offs[wid]; e < e1; ++e) {
    const int j  = csr_s[e];
    const float w = csr_w[e];
    float4 v = ((const float4*)(H + (long long)j * 128))[lane];
    acc.x += w * v.x; acc.y += w * v.y; acc.z += w * v.z; acc.w += w * v.w;
  }
  float4 bb = ((const float4*)bias)[lane];
  float4 o = make_float4(di * acc.x + bb.x, di * acc.y + bb.y,
                         di * acc.z + bb.z, di * acc.w + bb.w);
  if (do_relu) {
    o.x = fmaxf(o.x, 0.f); o.y = fmaxf(o.y, 0.f);
    o.z = fmaxf(o.z, 0.f); o.w = fmaxf(o.w, 0.f);
  }
  ((float4*)(out + (long long)wid * 128))[lane] = o;
}

// Final layer: 40 features. Lane l covers feat l; lanes 0-7 also feat 32+l.
__global__ __launch_bounds__(256) void gcn_agg40(
    const float* __restrict__ H, const float* __restrict__ bias,
    const float* __restrict__ dinv, const int* __restrict__ offs,
    const int* __restrict__ csr_s, const float* __restrict__ csr_w,
    float* __restrict__ out, int N_) {
  const int wid  = (int)((blockIdx.x * blockDim.x + threadIdx.x) >> 5);
  const int lane = threadIdx.x & 31;
  if (wid >= N_) return;
  const int f2 = 32 + (lane & 7);              // always in-range load
  const float di = dinv[wid];
  const float* hr = H + (long long)wid * 40;
  float a0 = di * hr[lane];
  float a1 = di * hr[f2];
  const int e1 = offs[wid + 1];
  for (int e = offs[wid]; e < e1; ++e) {
    const int j  = csr_s[e];
    const float w = csr_w[e];
    const float* hj = H + (long long)j * 40;
    a0 += w * hj[lane];
    a1 += w * hj[f2];
  }
  float* orow = out + (long long)wid * 40;
  orow[lane] = di * a0 + bias[lane];
  if (lane < 8) orow[f2] = di * a1 + bias[f2];
}

// ------------------------------- Launcher ----------------------------------
extern "C" void kernel_launch(void* const* d_in, const int* in_sizes, int n_in,
                              void* d_out, int out_size, void* d_ws, size_t ws_size,
                              hipStream_t stream) {
  const float* x  = (const float*)d_in[0];
  const float* W0 = (const float*)d_in[1];
  const float* b0 = (const float*)d_in[2];
  const float* W1 = (const float*)d_in[3];
  const float* b1 = (const float*)d_in[4];
  const float* W2 = (const float*)d_in[5];
  const float* b2 = (const float*)d_in[6];
  const int*   ei = (const int*)d_in[7];     // harness passes indices as int32

  const int N = in_sizes[0] / 128;
  const int E = in_sizes[7] / 2;
  const int* src = ei;                        // edge_index row 0
  const int* dst = ei + E;                    // edge_index row 1

  // Workspace carve-out (256B-aligned slices).
  char* p = (char*)d_ws;
  auto take = [&](size_t bytes) -> char* {
    char* r = p; p += (bytes + 255) & ~(size_t)255; return r;
  };
  float* dinv   = (float*)take(sizeof(float) * (size_t)N);
  int*   degI   = (int*)  take(sizeof(int)   * (size_t)N);
  int*   cursor = (int*)  take(sizeof(int)   * (size_t)N);
  int*   offs   = (int*)  take(sizeof(int)   * ((size_t)N + 1));
  const int NB  = (N + SCAN_B - 1) / SCAN_B;
  int*   parts  = (int*)  take(sizeof(int)   * (size_t)NB);
  int*   csr_s  = (int*)  take(sizeof(int)   * (size_t)E);
  float* csr_w  = (float*)take(sizeof(float) * (size_t)E);
  float* hA     = (float*)take(sizeof(float) * (size_t)N * 128);
  float* hB     = (float*)take(sizeof(float) * (size_t)N * 128);
  (void)ws_size; (void)n_in; (void)out_size;

  const int gN = (N + 255) / 256;
  const int gE = (E + 255) / 256;

  // --- degree / dinv / CSR (built every call; deterministic work) ---
  zero2_i32    <<<gN, 256, 0, stream>>>(degI, cursor, N);
  hist_deg     <<<gE, 256, 0, stream>>>(dst, degI, E);
  calc_dinv    <<<gN, 256, 0, stream>>>(degI, dinv, N);
  scan_block   <<<NB, SCAN_B, 0, stream>>>(degI, offs, parts, N);
  scan_partials<<<1, 32, 0, stream>>>(parts, NB);
  scan_add     <<<gN, 256, 0, stream>>>(offs, parts, N, E);
  csr_fill     <<<gE, 256, 0, stream>>>(src, dst, offs, cursor, dinv,
                                        csr_s, csr_w, E);

  // --- 3 GCN layers ---
  const int gGemm = (N + BM - 1) / BM;
  const int gAgg  = (N + 7) / 8;               // 8 waves (nodes) per block

  gemm_wmma_f32<128><<<gGemm, 256, 0, stream>>>(x, W0, hA, N);
  gcn_agg128        <<<gAgg, 256, 0, stream>>>(hA, b0, dinv, offs, csr_s, csr_w,
                                               hB, N, 1);
  gemm_wmma_f32<128><<<gGemm, 256, 0, stream>>>(hB, W1, hA, N);
  gcn_agg128        <<<gAgg, 256, 0, stream>>>(hA, b1, dinv, offs, csr_s, csr_w,
                                               hB, N, 1);
  gemm_wmma_f32<40> <<<gGemm, 256, 0, stream>>>(hB, W2, hA, N);
  gcn_agg40         <<<gAgg, 256, 0, stream>>>(hA, b2, dinv, offs, csr_s, csr_w,
                                               (float*)d_out, N);
}